// ActionSmoothingLoss_15788299780231
// MI455X (gfx1250) — compile-verified
//
#include <hip/hip_runtime.h>
#include <hip/hip_bf16.h>
#include <stdint.h>

// ---------------------------------------------------------------------------
// ActionSmoothingLoss on MI455X (gfx1250, wave32)
//
// loss = (1/W) * sum_{w,a} exp(t_wa) * (t_wa - x_a) * inv_n_a
//   x = segmented log_softmax(current_action)          [68]
//   t = segmented log_softmax(previous_actions rows)   [W,68]
//
// Memory-bound: 142.6 MB streamed once -> ~6us floor @ 23.3 TB/s.
// Design: async global->LDS tile staging (CDNA5 ASYNCcnt path), per-lane
// softmax from LDS, WMMA(16x16x4 f32 vs ones) wave reduction.
// ---------------------------------------------------------------------------

#define NSEG 6
#define ADIM 68
#define ROWS_PER_BLOCK 128
#define TPB 128
#define CHUNKS_PER_TILE ((ROWS_PER_BLOCK * ADIM) / 4)   /* 2176 float4 chunks */
#define CHUNKS_PER_THREAD (CHUNKS_PER_TILE / TPB)       /* 17 */

typedef float v2f __attribute__((ext_vector_type(2)));
typedef float v8f __attribute__((ext_vector_type(8)));

// ------------------------- prep: log_softmax(current) ----------------------
__global__ void asl_prep_kernel(const float* __restrict__ cur,
                                float* __restrict__ xbuf) {
    const int offs[NSEG] = {0, 3, 6, 10, 35, 60};
    const int lens[NSEG] = {3, 3, 4, 25, 25, 8};
    int seg = threadIdx.x;
    if (seg < NSEG) {
        int off = offs[seg], n = lens[seg];
        float m = cur[off];
        for (int i = 1; i < n; ++i) m = fmaxf(m, cur[off + i]);
        float s = 0.f;
        for (int i = 0; i < n; ++i) s += __expf(cur[off + i] - m);
        float logZ = m + __logf(s);
        for (int i = 0; i < n; ++i) xbuf[off + i] = cur[off + i] - logZ;
    }
}

// ------------------ per-segment KL term for one row (one lane) -------------
template <int N>
__device__ __forceinline__ float seg_term(const float* __restrict__ row,
                                          const float* __restrict__ xs,
                                          int off, float invn) {
    float v[N], e[N];
#pragma unroll
    for (int i = 0; i < N; ++i) v[i] = row[off + i];
    float m = v[0];
#pragma unroll
    for (int i = 1; i < N; ++i) m = fmaxf(m, v[i]);
    float s = 0.f;
#pragma unroll
    for (int i = 0; i < N; ++i) { e[i] = __expf(v[i] - m); s += e[i]; }
    float logZ = m + __logf(s);
    float rs   = __builtin_amdgcn_rcpf(s);
    float acc  = 0.f;
#pragma unroll
    for (int i = 0; i < N; ++i) {
        float t = v[i] - logZ;                 // log p
        acc = fmaf(e[i], t - xs[off + i], acc); // e * (t - x); scale by rs below
    }
    return acc * rs * invn;
}

// ----------------------------- main streaming kernel -----------------------
__global__ __launch_bounds__(TPB) void asl_main_kernel(
    const float* __restrict__ prev, const float* __restrict__ xbuf,
    float* __restrict__ partials, int W) {
    __shared__ float tile[ROWS_PER_BLOCK * ADIM];  // 34,816 B
    __shared__ float xs[ADIM];
    __shared__ float wsum[TPB / 32];

    const int tid = threadIdx.x;
    const long long row0       = (long long)blockIdx.x * ROWS_PER_BLOCK;
    const long long elem0      = row0 * ADIM;
    const long long totalElems = (long long)W * ADIM;

    if (tid < ADIM) xs[tid] = xbuf[tid];

    // --- Async DMA of the tile: global -> LDS, 16B per lane per issue.
    // Rows are 17 float4s each, tile chunks are coalesced across the block.
    unsigned ldsBase = (unsigned)(size_t)(&tile[0]);  // low 32 bits = LDS offset
#pragma unroll
    for (int k = 0; k < CHUNKS_PER_THREAD; ++k) {
        int c = tid + k * TPB;                        // float4 chunk index
        long long ge = elem0 + (long long)c * 4;
        if (ge + 4 <= totalElems) {
            const float* gp = prev + ge;
            unsigned loff = ldsBase + (unsigned)c * 16u;
            asm volatile("global_load_async_to_lds_b128 %0, %1, off"
                         :: "v"(loff), "v"(gp)
                         : "memory");
        }
    }
    asm volatile("s_wait_asynccnt 0" ::: "memory");
    __syncthreads();

    // --- One row per lane: segmented softmax + KL term.
    float acc = 0.f;
    if (row0 + tid < (long long)W) {
        const float* row = &tile[tid * ADIM];
        acc  = seg_term<3>(row, xs, 0, 1.f / 3.f);
        acc += seg_term<3>(row, xs, 3, 1.f / 3.f);
        acc += seg_term<4>(row, xs, 6, 0.25f);
        acc += seg_term<25>(row, xs, 10, 0.04f);
        acc += seg_term<25>(row, xs, 35, 0.04f);
        acc += seg_term<8>(row, xs, 60, 0.125f);
    }

    // --- Wave32 reduction through the matrix unit:
    // A[m,0]=acc(lane m), A[m,2]=acc(lane 16+m), B=ones  =>
    // D[m,n] = acc_m + acc_{16+m} (all n). Lane j<16 sums rows 0..7 over its
    // 8 D VGPRs, lane j>=16 sums rows 8..15; shfl_xor(16) completes the sum.
    float wave_total;
#if __has_builtin(__builtin_amdgcn_wmma_f32_16x16x4_f32)
    {
        v2f a; a.x = acc; a.y = 0.f;
        v2f b; b.x = 1.f; b.y = 1.f;
        v8f cz = {};
        v8f d = __builtin_amdgcn_wmma_f32_16x16x4_f32(
            /*neg_a=*/false, a, /*neg_b=*/false, b,
            /*c_mod=*/(short)0, cz, /*reuse_a=*/false, /*reuse_b=*/false);
        float t = ((d[0] + d[1]) + (d[2] + d[3])) + ((d[4] + d[5]) + (d[6] + d[7]));
        wave_total = t + __shfl_xor(t, 16, 32);
    }
#else
    {
        float t = acc;
        for (int o = 16; o > 0; o >>= 1) t += __shfl_xor(t, o, 32);
        wave_total = t;
    }
#endif

    int wave = tid >> 5, lane = tid & 31;
    if (lane == 0) wsum[wave] = wave_total;
    __syncthreads();
    if (tid == 0) {
        float s = 0.f;
#pragma unroll
        for (int i = 0; i < TPB / 32; ++i) s += wsum[i];
        partials[blockIdx.x] = s;
    }
}

// ----------------------------- final reduction -----------------------------
__global__ void asl_final_kernel(const float* __restrict__ partials, int nb,
                                 float invW, float* __restrict__ out) {
    float s = 0.f;
    for (int i = threadIdx.x; i < nb; i += 32) s += partials[i];
    for (int o = 16; o > 0; o >>= 1) s += __shfl_xor(s, o, 32);
    if (threadIdx.x == 0) out[0] = s * invW;
}

// ---------------------------------------------------------------------------
extern "C" void kernel_launch(void* const* d_in, const int* in_sizes, int n_in,
                              void* d_out, int out_size, void* d_ws, size_t ws_size,
                              hipStream_t stream) {
    const float* cur  = (const float*)d_in[0];   // [68]
    const float* prev = (const float*)d_in[1];   // [W, 68]
    float* out = (float*)d_out;

    float* xbuf     = (float*)d_ws;              // 68 floats
    float* partials = (float*)d_ws + 256;        // nb floats (<= 4096)

    int W  = in_sizes[1] / ADIM;
    int nb = (W + ROWS_PER_BLOCK - 1) / ROWS_PER_BLOCK;

    asl_prep_kernel<<<1, 32, 0, stream>>>(cur, xbuf);
    asl_main_kernel<<<nb, TPB, 0, stream>>>(prev, xbuf, partials, W);
    asl_final_kernel<<<1, 32, 0, stream>>>(partials, nb, 1.0f / (float)W, out);
}